// FastMultiEMAMixture_4870492914162
// MI455X (gfx1250) — compile-verified
//
#include <hip/hip_runtime.h>
#include <hip/hip_bf16.h>

// Problem constants (match reference)
#define KK    4      // number of EMAs
#define CCH   512    // channels
#define BB    16     // batch
#define TTOT  4096   // timesteps
#define NSEG  16     // supersegments (blocked scan)
#define TS    256    // TTOT / NSEG
#define LSEG  16     // inner segment length = WMMA M
#define XPITCH 20    // LDS [t][b][c] pitch in floats: 80B rows (16B aligned),
                     // b-stride 20 -> bank-conflict-free operand reads

typedef __attribute__((ext_vector_type(2))) float v2f;
typedef __attribute__((ext_vector_type(8))) float v8f;
typedef __attribute__((__vector_size__(4 * sizeof(int)))) int i32x4;

#if __has_builtin(__builtin_amdgcn_global_load_async_to_lds_b128) && \
    __has_builtin(__builtin_amdgcn_global_store_async_from_lds_b128)
#define USE_ASYNC_LDS 1
#else
#define USE_ASYNC_LDS 0
#endif

// Global-side: generic typed vector pointer (per clang's diagnosed signature).
#define GP(p) ((i32x4*)(p))
// LDS-side: typed addrspace(3) pointer (converts to AS3 void* if needed).
#define LP(p) ((__attribute__((address_space(3))) i32x4*)(p))

#if USE_ASYNC_LDS
__device__ __forceinline__ void async_wait0() {
#if __has_builtin(__builtin_amdgcn_s_wait_asynccnt)
  __builtin_amdgcn_s_wait_asynccnt(0);
#else
  asm volatile("s_wait_asynccnt 0x0" ::: "memory");
#endif
}
#endif

__device__ __forceinline__ float sig_clamp(float l) {
  float s = 1.0f / (1.0f + __expf(-l));
  return fminf(fmaxf(s, 1e-4f), 1.0f - 1e-4f);
}

__device__ __forceinline__ float powi(float a, int e) {
  float p = 1.0f;
  for (int i = 0; i < e; ++i) p *= a;
  return p;
}

// ---------------------------------------------------------------------------
// Pass 1: per (superseg, b, c) zero-init local EMA end-states. Coalesced in c.
// ---------------------------------------------------------------------------
__global__ __launch_bounds__(256) void ema_partials(
    const float* __restrict__ x, const float* __restrict__ logit_alpha,
    float* __restrict__ Ls) {
  int g = blockIdx.x * blockDim.x + threadIdx.x;   // [0, NSEG*BB*CCH)
  int c = g % CCH;
  int b = (g / CCH) % BB;
  int s = g / (CCH * BB);

  float a[KK], y[KK];
#pragma unroll
  for (int k = 0; k < KK; ++k) a[k] = sig_clamp(logit_alpha[k * CCH + c]);

  const float* xp = x + ((size_t)b * TTOT + (size_t)s * TS) * CCH + c;
  int t0;
  if (s == 0) {                      // y[0] = x[0] boundary
    float x0 = xp[0];
#pragma unroll
    for (int k = 0; k < KK; ++k) y[k] = x0;
    t0 = 1;
  } else {
#pragma unroll
    for (int k = 0; k < KK; ++k) y[k] = 0.0f;
    t0 = 0;
  }
#pragma unroll 4
  for (int t = t0; t < TS; ++t) {
    float xv = xp[(size_t)t * CCH];
#pragma unroll
    for (int k = 0; k < KK; ++k) y[k] = fmaf(a[k], y[k] - xv, xv);  // a*y+(1-a)*x
  }
#pragma unroll
  for (int k = 0; k < KK; ++k)
    Ls[(((size_t)s * KK + k) * BB + b) * CCH + c] = y[k];
}

// ---------------------------------------------------------------------------
// Pass 2: serial combine over supersegments -> carry-in state per superseg.
// ---------------------------------------------------------------------------
__global__ __launch_bounds__(256) void ema_combine(
    const float* __restrict__ logit_alpha, const float* __restrict__ Ls,
    float* __restrict__ Carry) {
  int g = blockIdx.x * blockDim.x + threadIdx.x;   // [0, KK*BB*CCH)
  int c = g % CCH;
  int b = (g / CCH) % BB;
  int k = g / (CCH * BB);

  float a = sig_clamp(logit_alpha[k * CCH + c]);
  float F = a;
#pragma unroll
  for (int i = 0; i < 8; ++i) F = F * F;           // a^256 == a^TS

  float E = Ls[(((size_t)0 * KK + k) * BB + b) * CCH + c];
  for (int s = 1; s < NSEG; ++s) {
    Carry[(((size_t)s * KK + k) * BB + b) * CCH + c] = E;
    E = fmaf(F, E, Ls[(((size_t)s * KK + k) * BB + b) * CCH + c]);
  }
}

// ---------------------------------------------------------------------------
// Pass 3: WMMA scan-as-matmul with async-LDS DMA tile staging.
// Workgroup = 16 waves; wave w owns channel c = c0+w; block y = superseg.
// ---------------------------------------------------------------------------
__global__ __launch_bounds__(512) void ema_wmma(
    const float* __restrict__ x, const float* __restrict__ logit_alpha,
    const float* __restrict__ mix_logits, const float* __restrict__ Carry,
    float* __restrict__ out) {
  __shared__ __align__(16) float xs[LSEG * 16 * XPITCH];   // [t][b][c]
  __shared__ __align__(16) float os[LSEG * 16 * XPITCH];   // trend staging

  const int tid  = threadIdx.x;
  const int w    = tid >> 5;             // wave id = channel within tile
  const int lane = tid & 31;
  const int ln   = lane & 15;
  const bool lo  = lane < 16;
  const int q0   = lo ? 0 : 2;           // f32 A/B layout: K 0,1 | 2,3 split
  const int c0   = blockIdx.x * 16;
  const int c    = c0 + w;
  const int s    = blockIdx.y;

  // per-k alpha + softmax mix (wave-uniform, computed redundantly per lane)
  float a[KK], mix[KK];
  float lmax = -1e30f;
#pragma unroll
  for (int k = 0; k < KK; ++k) {
    a[k]   = sig_clamp(logit_alpha[k * CCH + c]);
    mix[k] = mix_logits[c * KK + k];
    lmax   = fmaxf(lmax, mix[k]);
  }
  float msum = 0.0f;
#pragma unroll
  for (int k = 0; k < KK; ++k) { mix[k] = __expf(mix[k] - lmax); msum += mix[k]; }
  float minv = 1.0f / msum;
#pragma unroll
  for (int k = 0; k < KK; ++k) mix[k] *= minv;

  // Build A operands once (reused for all 16 inner segments).
  // W[m][j] = (j<=m) ? (1-a)*a^(m-j) : 0 ; carry column: a^(m+1).
  v2f Aop[KK][4], Acar[KK];
#pragma unroll
  for (int k = 0; k < KK; ++k) {
    float ak = a[k], om = 1.0f - ak;
#pragma unroll
    for (int kk = 0; kk < 4; ++kk) {
      int j0 = 4 * kk + q0;
      Aop[k][kk].x = (j0     <= ln) ? om * powi(ak, ln - j0)     : 0.0f;
      Aop[k][kk].y = (j0 + 1 <= ln) ? om * powi(ak, ln - j0 - 1) : 0.0f;
    }
    Acar[k].x = lo ? powi(ak, ln + 1) : 0.0f;  // K=0 column only
    Acar[k].y = 0.0f;
  }

  // Initial carries (b = ln, valid in lanes 0..15).
  float cin[KK];
  if (lo) {
    if (s == 0) {     // carry = x0 makes y[0] = a*x0 + (1-a)*x0 = x0
      float x0 = x[((size_t)ln * TTOT) * CCH + c];
#pragma unroll
      for (int k = 0; k < KK; ++k) cin[k] = x0;
    } else {
#pragma unroll
      for (int k = 0; k < KK; ++k)
        cin[k] = Carry[(((size_t)s * KK + k) * BB + ln) * CCH + c];
    }
  } else {
#pragma unroll
    for (int k = 0; k < KK; ++k) cin[k] = 0.0f;
  }

  // Cooperative tile geometry: one b128 (4 floats of c) per thread per round.
  const int ccq = (tid & 3) * 4;
  const int ttc = (tid >> 2) & 15;
  const int bb0 = tid >> 6;                               // 0..7
  const int lofA = (ttc * 16 + bb0) * XPITCH + ccq;       // b = bb0
  const int lofB = lofA + 8 * XPITCH;                     // b = bb0 + 8
  const size_t gofR = (size_t)8 * TTOT * CCH;             // +8 in b
  const size_t gof0 =
      ((size_t)bb0 * TTOT + (size_t)s * TS + ttc) * CCH + c0 + ccq;
  const float* gx   = x   + gof0;
  float*       gout = out + gof0;

  for (int i = 0; i < TS / LSEG; ++i) {
    // ---- Stage x tile into LDS (async DMA, no VGPR round-trip) ----
#if USE_ASYNC_LDS
    __builtin_amdgcn_global_load_async_to_lds_b128(GP(gx),        LP(xs + lofA), 0, 0);
    __builtin_amdgcn_global_load_async_to_lds_b128(GP(gx + gofR), LP(xs + lofB), 0, 0);
    async_wait0();   // x tile ready; also drains previous segment's out-stores
#else
    float4 va = *(const float4*)(gx);
    float4 vb = *(const float4*)(gx + gofR);
    *(float4*)(xs + lofA) = va;
    *(float4*)(xs + lofB) = vb;
#endif
    if (i + 1 < TS / LSEG)
      __builtin_prefetch(gx + (size_t)LSEG * CCH, 0, 3);
    __syncthreads();

    // ---- Per-wave WMMA compute ----
    v2f Xb[4];
#pragma unroll
    for (int kk = 0; kk < 4; ++kk) {
      Xb[kk].x = xs[((4 * kk + q0)     * 16 + ln) * XPITCH + w];
      Xb[kk].y = xs[((4 * kk + q0 + 1) * 16 + ln) * XPITCH + w];
    }

    float tr[8] = {0, 0, 0, 0, 0, 0, 0, 0};
#pragma unroll
    for (int k = 0; k < KK; ++k) {
      v8f acc = {0, 0, 0, 0, 0, 0, 0, 0};
      v2f bc;
      bc.x = lo ? cin[k] : 0.0f;   // carry row K=0; K=2 row (upper lanes) = 0
      bc.y = 0.0f;
      acc = __builtin_amdgcn_wmma_f32_16x16x4_f32(
          false, Acar[k], false, bc, (short)0, acc, false, false);
#pragma unroll
      for (int kk = 0; kk < 4; ++kk)
        acc = __builtin_amdgcn_wmma_f32_16x16x4_f32(
            false, Aop[k][kk], false, Xb[kk], (short)0, acc, false, false);

      // New carry = y row M=15: VGPR 7, lanes 16..31 (N = lane-16).
      float last = acc[7];
      cin[k] = __shfl(last, 16 + ln, 32);

#pragma unroll
      for (int r = 0; r < 8; ++r) tr[r] = fmaf(mix[k], acc[r], tr[r]);
    }

    // Stage trend to LDS [t][b][c]: lane holds (t = r or r+8, b = ln), c = w.
#pragma unroll
    for (int r = 0; r < 8; ++r) {
      int tt = r + (lo ? 0 : 8);
      os[(tt * 16 + ln) * XPITCH + w] = tr[r];
    }
    __syncthreads();

    // ---- Drain trend tile to global (async DMA from LDS) ----
#if USE_ASYNC_LDS
    __builtin_amdgcn_global_store_async_from_lds_b128(GP(gout),        LP(os + lofA), 0, 0);
    __builtin_amdgcn_global_store_async_from_lds_b128(GP(gout + gofR), LP(os + lofB), 0, 0);
#else
    float4 oa = *(const float4*)(os + lofA);
    float4 ob = *(const float4*)(os + lofB);
    *(float4*)(gout)        = oa;
    *(float4*)(gout + gofR) = ob;
#endif

    gx   += (size_t)LSEG * CCH;
    gout += (size_t)LSEG * CCH;
  }
  // s_endpgm's implicit wait-idle drains the final async stores.
}

// ---------------------------------------------------------------------------
extern "C" void kernel_launch(void* const* d_in, const int* in_sizes, int n_in,
                              void* d_out, int out_size, void* d_ws, size_t ws_size,
                              hipStream_t stream) {
  (void)in_sizes; (void)n_in; (void)out_size; (void)ws_size;
  const float* x  = (const float*)d_in[0];   // [B,T,C] f32
  const float* la = (const float*)d_in[1];   // [K,C]   f32
  const float* ml = (const float*)d_in[2];   // [C,K]   f32
  float* out = (float*)d_out;                // [B,T,C] f32

  float* Ls    = (float*)d_ws;                          // NSEG*K*B*C floats (2 MB)
  float* Carry = Ls + (size_t)NSEG * KK * BB * CCH;     // NSEG*K*B*C floats (2 MB)

  ema_partials<<<dim3((NSEG * BB * CCH) / 256), dim3(256), 0, stream>>>(x, la, Ls);
  ema_combine<<<dim3((KK * BB * CCH) / 256), dim3(256), 0, stream>>>(la, Ls, Carry);
  ema_wmma<<<dim3(CCH / 16, NSEG), dim3(512), 0, stream>>>(x, la, ml, Carry, out);
}